// CapsuleFC_49452253447032
// MI455X (gfx1250) — compile-verified
//
#include <hip/hip_runtime.h>

// CapsuleFC routing on MI455X (gfx1250).
// Memory-bound: w is 256MB, streamed once per pass (2 passes = 512MB -> ~22us
// floor @ 23.3 TB/s HBM). FP32 WMMA (V_WMMA_F32_16X16X4_F32) keeps reference
// numerics; all small tensors (pose/ncv/qk/out) are L2-resident (192MB L2).

typedef __attribute__((ext_vector_type(2))) float v2f;
typedef __attribute__((ext_vector_type(8))) float v8f;

#define B_  32
#define N_  512
#define A_  32
#define M_  128
#define D_  32
#define MD_ (M_ * D_)            // 4096
#define SCALE_ 0.17677669529663687f

static __device__ __forceinline__ v8f wmma_f32(v2f a, v2f b, v8f c) {
  // 8-arg pattern: (neg_a, A, neg_b, B, c_mod, C, reuse_a, reuse_b)
  return __builtin_amdgcn_wmma_f32_16x16x4_f32(false, a, false, b, (short)0, c,
                                               false, false);
}

// ---------------------------------------------------------------------------
// Kernel 1: one block per n. Computes logits[:, n, :] via WMMA votes,
// then softmax over m, modulate by next_act, renormalize -> qk[:, n, :].
// GEMM: A = pose[:,n,:] [32b x 32a], B = w[n] [32a x 4096(m,d)].
// ---------------------------------------------------------------------------
__global__ __launch_bounds__(256) void caps_logits_qk(
    const float* __restrict__ pose, const float* __restrict__ ncv,
    const float* __restrict__ na, const float* __restrict__ w,
    float* __restrict__ qk_out) {
  const int n    = blockIdx.x;
  const int tid  = threadIdx.x;
  const int lane = tid & 31;          // wave32
  const int wv   = tid >> 5;          // 8 waves
  const int l16  = lane & 15;
  const int khalf = (lane < 16) ? 0 : 2;  // K offset per A/B VGPR layout

  __shared__ float vote[32 * 260];    // [32 b][256 cols] padded stride
  __shared__ float lg[32 * 128];      // logits -> softmax workspace

  // Preload A operand: areg[rowtile][kstep] = pose[b, n, 4k+khalf .. +1]
  v2f areg[2][8];
  for (int rt = 0; rt < 2; ++rt) {
    const int b = l16 + 16 * rt;
    const float* pp = pose + ((size_t)b * N_ + n) * A_;
    for (int k = 0; k < 8; ++k) areg[rt][k] = *(const v2f*)(pp + 4 * k + khalf);
  }

  for (int mt = 0; mt < 16; ++mt) {       // m-tiles of 8 m (256 cols)
    for (int j = 0; j < 2; ++j) {         // 2 col-tiles per wave (16 total)
      const int ct  = wv * 2 + j;
      const int col = mt * 256 + ct * 16 + l16;
      const float* wbase = w + (size_t)n * A_ * MD_ + col;
      __builtin_prefetch(wbase + 256, 0, 0);  // next m-tile, same rows
      v8f acc0 = {}, acc1 = {};
      for (int k = 0; k < 8; ++k) {         // K = 32 in steps of 4
        const int a = 4 * k + khalf;
        const float* wp = wbase + (size_t)a * MD_;
        v2f bv; bv.x = wp[0]; bv.y = wp[MD_];   // rows a, a+1
        acc0 = wmma_f32(areg[0][k], bv, acc0);
        acc1 = wmma_f32(areg[1][k], bv, acc1);
      }
      const int cbase = ct * 16 + l16;
      const int rbase = (lane < 16) ? 0 : 8;
      for (int i = 0; i < 8; ++i) {           // C layout: vgpr i = rows i/i+8
        vote[(i + rbase) * 260 + cbase]      = acc0[i];
        vote[(i + rbase + 16) * 260 + cbase] = acc1[i];
      }
    }
    __syncthreads();
    // Dot with ncv over d: one thread per (b, m) pair
    {
      const int b  = tid & 31;
      const int mi = tid >> 5;                 // 0..7
      const int m  = mt * 8 + mi;
      const float* vp = vote + b * 260 + mi * 32;
      const float* cp = ncv + ((size_t)b * M_ + m) * D_;
      float s = 0.f;
      for (int d = 0; d < 32; ++d) s += vp[d] * cp[d];
      lg[b * 128 + m] = s * SCALE_;
    }
    __syncthreads();
  }

  // Softmax over m, modulate by next_act, renormalize (one thread per b).
  if (tid < 32) {
    const int b = tid;
    float* row = lg + b * 128;
    float mx = -3.4e38f;
    for (int m = 0; m < 128; ++m) mx = fmaxf(mx, row[m]);
    float s = 0.f;
    for (int m = 0; m < 128; ++m) { float e = __expf(row[m] - mx); row[m] = e; s += e; }
    const float inv = 1.f / s;
    const float* nap = na + b * 128;
    float s2 = 0.f;
    for (int m = 0; m < 128; ++m) { float v = row[m] * inv * nap[m]; row[m] = v; s2 += v; }
    const float inv2 = 1.f / (s2 + 1e-10f);
    float* qp = qk_out + ((size_t)b * N_ + n) * M_;
    for (int m = 0; m < 128; ++m) qp[m] = row[m] * inv2;
  }
}

// ---------------------------------------------------------------------------
// Kernel 2: grid (m=128, dhalf=2). out[:, m, dh*16..+16] =
//   [32b x 16384(n,a)] @ [16384 x 16d], A built on the fly as qk'*pose.
// Waves split the n-range; per-wave LDS partials summed in fixed order
// (deterministic, no atomics).
// ---------------------------------------------------------------------------
__global__ __launch_bounds__(256) void caps_out(
    const float* __restrict__ pose, const float* __restrict__ ca,
    const float* __restrict__ w, const float* __restrict__ qk,
    float* __restrict__ out) {
  const int m    = blockIdx.x;
  const int dh   = blockIdx.y;
  const int tid  = threadIdx.x;
  const int lane = tid & 31;
  const int wv   = tid >> 5;
  const int l16  = lane & 15;
  const int khalf = (lane < 16) ? 0 : 2;
  const int dcol = dh * 16 + l16;

  __shared__ float red[8][32 * 17];   // per-wave partials, padded stride

  v8f acc0 = {}, acc1 = {};
  const int b0 = l16, b1 = l16 + 16;

  for (int n = wv * 64; n < wv * 64 + 64; ++n) {
    const float q0 = qk[((size_t)b0 * N_ + n) * M_ + m] * ca[(size_t)b0 * N_ + n];
    const float q1 = qk[((size_t)b1 * N_ + n) * M_ + m] * ca[(size_t)b1 * N_ + n];
    const float* pp0 = pose + ((size_t)b0 * N_ + n) * A_;
    const float* pp1 = pose + ((size_t)b1 * N_ + n) * A_;
    const float* wbase = w + (size_t)n * A_ * MD_ + m * D_ + dcol;
    __builtin_prefetch(wbase + (size_t)2 * A_ * MD_, 0, 0);  // two n ahead
    for (int k = 0; k < 8; ++k) {
      const int a = 4 * k + khalf;
      const v2f p0 = *(const v2f*)(pp0 + a);
      const v2f p1 = *(const v2f*)(pp1 + a);
      v2f av0; av0.x = q0 * p0.x; av0.y = q0 * p0.y;
      v2f av1; av1.x = q1 * p1.x; av1.y = q1 * p1.y;
      const float* wp = wbase + (size_t)a * MD_;
      v2f bv; bv.x = wp[0]; bv.y = wp[MD_];
      acc0 = wmma_f32(av0, bv, acc0);
      acc1 = wmma_f32(av1, bv, acc1);
    }
  }

  const int rbase = (lane < 16) ? 0 : 8;
  for (int i = 0; i < 8; ++i) {
    red[wv][(i + rbase) * 17 + l16]      = acc0[i];
    red[wv][(i + rbase + 16) * 17 + l16] = acc1[i];
  }
  __syncthreads();
  for (int e = tid; e < 512; e += 256) {
    const int b = e >> 4, dc = e & 15;
    float s = 0.f;
    for (int v = 0; v < 8; ++v) s += red[v][b * 17 + dc];  // fixed order
    out[(size_t)b * MD_ + m * D_ + dh * 16 + dc] = s;
  }
}

// ---------------------------------------------------------------------------
extern "C" void kernel_launch(void* const* d_in, const int* in_sizes, int n_in,
                              void* d_out, int out_size, void* d_ws,
                              size_t ws_size, hipStream_t stream) {
  (void)in_sizes; (void)n_in; (void)out_size; (void)d_ws; (void)ws_size;
  const float* pose = (const float*)d_in[0];   // [B,N,A]
  const float* ca   = (const float*)d_in[1];   // [B,N]
  const float* ncv  = (const float*)d_in[2];   // [B,M,D]
  const float* na   = (const float*)d_in[3];   // [B,M]
  const float* w    = (const float*)d_in[4];   // [N,A,M,D]
  // d_in[5] = num_iter (1): single routing pass, as in the reference.

  float* out    = (float*)d_out;               // [B,M,D]  = 131072
  float* na_out = out + (size_t)B_ * M_ * D_;  // [B,M]    = 4096
  float* qk_out = na_out + (size_t)B_ * M_;    // [B,N,M]  = 2097152

  caps_logits_qk<<<dim3(N_), dim3(256), 0, stream>>>(pose, ncv, na, w, qk_out);
  hipMemcpyAsync(na_out, na, (size_t)B_ * M_ * sizeof(float),
                 hipMemcpyDeviceToDevice, stream);
  caps_out<<<dim3(M_, 2), dim3(256), 0, stream>>>(pose, ca, w, qk_out, out);
}